// QuantumKernelClassifier_87428354277466
// MI455X (gfx1250) — compile-verified
//
#include <hip/hip_runtime.h>
#include <math.h>

typedef __attribute__((ext_vector_type(2))) float v2f;
typedef __attribute__((ext_vector_type(8))) float v8f;
typedef __attribute__((ext_vector_type(4))) int   v4i;

#define WAVES          2
#define ROWS_PER_WAVE  16
#define TK             128   // K-chunk staged in LDS per wave-tile
#define NCHUNK         (DPROJ / TK)                    // 8
#define A_ROW_STRIDE   132   // floats; 132%64==4 -> ds_load_b64 fragments hit 64 distinct banks
#define A_BUF_STRIDE   (ROWS_PER_WAVE * A_ROW_STRIDE)  // one tile buffer, floats
#define W_ROW_STRIDE   1032  // floats; 1032%64==8 -> W rows land in distinct bank groups
#define DPROJ          1024
#define NQ             4

#define AS1 __attribute__((address_space(1)))
#define AS3 __attribute__((address_space(3)))

#if defined(__AMDGCN__) && __has_builtin(__builtin_amdgcn_global_load_async_to_lds_b128) && \
    __has_builtin(__builtin_amdgcn_s_wait_asynccnt)
#define USE_ASYNC_LDS 1
#else
#define USE_ASYNC_LDS 0
#endif

#if defined(__AMDGCN__) && __has_builtin(__builtin_amdgcn_s_wait_dscnt)
#define WAIT_DS0() __builtin_amdgcn_s_wait_dscnt(0)
#else
#define WAIT_DS0() __syncthreads()
#endif

#define CFENCE() asm volatile("" ::: "memory")

__global__ __launch_bounds__(WAVES * 32)
void qkc_wmma_kernel(const float* __restrict__ features,
                     const float* __restrict__ W_proj,
                     const float* __restrict__ b_proj,
                     const float* __restrict__ ref_vec,
                     const float* __restrict__ W_cls,
                     const float* __restrict__ b_cls,
                     float* __restrict__ out)
{
    __shared__ float ldsW[NQ * W_ROW_STRIDE];                 // 16.1 KB
    __shared__ float ldsA[WAVES * 2 * A_BUF_STRIDE];          // 33.0 KB (double-buffered)
    __shared__ float ldsC[WAVES * 256];                       //  2.0 KB
    __shared__ float ldsZ[TK + 8];                            // zero pad for N>=4 B lanes

    const int tid  = threadIdx.x;
    const int wave = tid >> 5;
    const int lane = tid & 31;
    const int half = lane >> 4;        // 0: lanes 0-15, 1: lanes 16-31
    const int koff = half * 2;         // upper half-wave carries K+2,K+3
    const int nIdx = lane & 15;        // M for A-frag, N for B/C-frag

    // Stage W_proj (4 x 1024) into padded LDS rows; zero the padding region.
    for (int i = tid; i < NQ * DPROJ; i += WAVES * 32) {
        const int n = i >> 10;
        const int k = i & (DPROJ - 1);
        ldsW[n * W_ROW_STRIDE + k] = W_proj[n * DPROJ + k];
    }
    for (int i = tid; i < TK + 8; i += WAVES * 32)
        ldsZ[i] = 0.0f;

    const int rowBase = blockIdx.x * (WAVES * ROWS_PER_WAVE) + wave * ROWS_PER_WAVE;
    float* aBuf0 = &ldsA[wave * 2 * A_BUF_STRIDE];

    // Issue one chunk's async DMA (16 x b128 per lane-row): global -> LDS, ASYNCcnt-tracked.
    auto issue_dma = [&](int kc, float* buf) {
#if USE_ASYNC_LDS
        #pragma unroll
        for (int r = 0; r < ROWS_PER_WAVE; ++r) {
            const float* src = features + (size_t)(rowBase + r) * DPROJ + kc + lane * 4;
            float* dst = buf + r * A_ROW_STRIDE + lane * 4;
            if (kc + TK < DPROJ)
                __builtin_prefetch(src + TK, 0, 3);   // warm GL2 one more chunk ahead
            __builtin_amdgcn_global_load_async_to_lds_b128(
                (AS1 v4i*)src, (AS3 v4i*)dst, 0, 0);
        }
#else
        float4 stage[ROWS_PER_WAVE];
        #pragma unroll
        for (int r = 0; r < ROWS_PER_WAVE; ++r)
            stage[r] = *(const float4*)(features + (size_t)(rowBase + r) * DPROJ + kc + lane * 4);
        #pragma unroll
        for (int r = 0; r < ROWS_PER_WAVE; ++r)
            *(float4*)(buf + r * A_ROW_STRIDE + lane * 4) = stage[r];
#endif
    };

    v8f acc = {};

    __syncthreads();          // ldsW / ldsZ visible to all waves

    issue_dma(0, aBuf0);      // prologue: fill buffer 0

    for (int c = 0; c < NCHUNK; ++c) {
        const int kc = c * TK;
        float* cur = aBuf0 + (c & 1) * A_BUF_STRIDE;
        float* nxt = aBuf0 + ((c + 1) & 1) * A_BUF_STRIDE;

        if (c + 1 < NCHUNK) {
            issue_dma(kc + TK, nxt);                     // overlap next DMA with this compute
#if USE_ASYNC_LDS
            __builtin_amdgcn_s_wait_asynccnt(ROWS_PER_WAVE);  // chunk c done; c+1 in flight
#endif
        } else {
#if USE_ASYNC_LDS
            __builtin_amdgcn_s_wait_asynccnt(0);
#endif
        }
#if !USE_ASYNC_LDS
        WAIT_DS0();
#endif
        CFENCE();   // keep ds_loads below the counter wait

        // B-fragment base: lanes with N<4 walk their W row; others walk a zero region.
        const float* bBase = (nIdx < NQ) ? &ldsW[nIdx * W_ROW_STRIDE + kc] : &ldsZ[0];

        // 32 x V_WMMA_F32_16X16X4_F32, fragments from LDS.
        #pragma unroll
        for (int kk = 0; kk < TK / 4; ++kk) {
            v2f a = *(const v2f*)(cur + nIdx * A_ROW_STRIDE + kk * 4 + koff);
            v2f b = *(const v2f*)(bBase + kk * 4 + koff);
            acc = __builtin_amdgcn_wmma_f32_16x16x4_f32(
                /*neg_a=*/false, a, /*neg_b=*/false, b,
                /*c_mod=*/(short)0, acc, /*reuse_a=*/false, /*reuse_b=*/false);
        }

        // Drain this wave's LDS reads before `cur` is DMA-overwritten at chunk c+1.
        CFENCE();
        WAIT_DS0();
        CFENCE();
    }

    // Scatter C fragment (VGPR j: lanes 0-15 -> M=j, lanes 16-31 -> M=j+8, N=lane&15)
    // through LDS so each row's 4 valid dot products land in one lane.
    float* cS = &ldsC[wave * 256];
    #pragma unroll
    for (int j = 0; j < 8; ++j) {
        cS[(j + half * 8) * 16 + nIdx] = acc[j];
    }
    __syncthreads();   // cross-lane scatter/gather: force compiler + HW ordering

    if (lane < ROWS_PER_WAVE) {
        const int row = lane;
        float ov = 1.0f;
        #pragma unroll
        for (int n = 0; n < NQ; ++n) {
            const float y = tanhf(cS[row * 16 + n] + b_proj[n]);
            const float u = tanhf(ref_vec[n]);
            ov *= cosf(0.5f * (y - u));   // per-qubit overlap is real: cos((y-u)/2)
        }
        const float fid = ov * ov;        // |Π cos|^2
        out[rowBase + row] = fid * W_cls[0] + b_cls[0];
    }
}

extern "C" void kernel_launch(void* const* d_in, const int* in_sizes, int n_in,
                              void* d_out, int out_size, void* d_ws, size_t ws_size,
                              hipStream_t stream) {
    const float* features = (const float*)d_in[0];
    const float* W_proj   = (const float*)d_in[1];
    const float* b_proj   = (const float*)d_in[2];
    const float* ref_vec  = (const float*)d_in[3];
    const float* W_cls    = (const float*)d_in[4];
    const float* b_cls    = (const float*)d_in[5];
    float* out = (float*)d_out;

    const int B = in_sizes[0] / DPROJ;                     // 65536
    const int rowsPerBlock = WAVES * ROWS_PER_WAVE;        // 32
    dim3 grid((B + rowsPerBlock - 1) / rowsPerBlock);      // 2048
    dim3 block(WAVES * 32);                                // 64 (2 wave32)

    qkc_wmma_kernel<<<grid, block, 0, stream>>>(features, W_proj, b_proj,
                                                ref_vec, W_cls, b_cls, out);
}